// CacheModel_19748259627242
// MI455X (gfx1250) — compile-verified
//
#include <hip/hip_runtime.h>
#include <hip/hip_bf16.h>
#include <math.h>

#define D_FEAT 2048
#define N_MEM 200000
#define N_CLASSES 1000
#define THETA 5.0f

typedef __attribute__((ext_vector_type(2))) float v2f;
typedef __attribute__((ext_vector_type(8))) float v8f;

// ---------------------------------------------------------------------------
// Kernel 1: ||x|| reduction, vn = theta * x / ||x||, zero class partials
// ---------------------------------------------------------------------------
__global__ __launch_bounds__(256) void prep_kernel(const float* __restrict__ x,
                                                   float* __restrict__ vn,
                                                   float* __restrict__ p_partial) {
    __shared__ float red[256];
    const int t = threadIdx.x;
    float s = 0.0f;
    for (int d = t; d < D_FEAT; d += 256) {
        float v = x[d];
        s += v * v;
    }
    red[t] = s;
    __syncthreads();
    for (int off = 128; off > 0; off >>= 1) {
        if (t < off) red[t] += red[t + off];
        __syncthreads();
    }
    const float scale = THETA / sqrtf(red[0]);
    for (int d = t; d < D_FEAT; d += 256) vn[d] = x[d] * scale;
    for (int c = t; c < 1024; c += 256) p_partial[c] = 0.0f;
}

// ---------------------------------------------------------------------------
// Kernel 2: sims[j] = exp( vn . K[:,j] ) via V_WMMA_F32_16X16X4_F32.
// A 16x4 fragment: row 0 = vn[k0..k0+3], rows 1..15 = 0.
//   A layout (ISA 7.12.2): A[m][k] -> lane = m + 16*(k>>1), vgpr = k&1.
//   So lane 0 holds {vn[k0], vn[k0+1]}, lane 16 holds {vn[k0+2], vn[k0+3]}.
// B 4x16 fragment (row k striped across lanes within a VGPR, mirroring A):
//   B[k][n] -> lane = n + 16*(k>>1), vgpr = k&1.
// D row 0 lands in acc[0] of lanes 0..15 (N = lane).
// Each wave covers 32 columns (two 16-wide tiles) so the 4 K-rows touched per
// k-step are read as contiguous 128B spans.
// ---------------------------------------------------------------------------
__global__ __launch_bounds__(256) void sims_wmma_kernel(const float* __restrict__ K,
                                                        const float* __restrict__ vn,
                                                        float* __restrict__ sims) {
    const int lane   = threadIdx.x & 31;
    const int wave   = threadIdx.x >> 5;
    const int laneLo = lane & 15;
    const int hi     = lane >> 4;                  // 0: k0/k0+1 rows, 1: k0+2/k0+3 rows
    const int jbase  = blockIdx.x * 256 + wave * 32;

    const int j0  = jbase + laneLo;                // tile 0 column for this lane
    const int j1  = jbase + 16 + laneLo;           // tile 1 column
    const int j0c = (j0 < N_MEM) ? j0 : (N_MEM - 1);
    const int j1c = (j1 < N_MEM) ? j1 : (N_MEM - 1);

    const float mask = (laneLo == 0) ? 1.0f : 0.0f;

    v8f acc0 = {};
    v8f acc1 = {};

    for (int k0 = 0; k0 < D_FEAT; k0 += 4) {
        const int kr = k0 + 2 * hi;                // first K-row handled by this half-wave

        // A fragment: broadcast query values, zeroed outside row 0
        float2 av = *(const float2*)(vn + kr);     // vn stays L2/L0 resident (8 KB)
        v2f a;
        a[0] = av.x * mask;
        a[1] = av.y * mask;

        const float* __restrict__ row0 = K + (size_t)kr * N_MEM;
        const float* __restrict__ row1 = row0 + N_MEM;

        v2f b0, b1;
        b0[0] = row0[j0c];
        b0[1] = row1[j0c];
        b1[0] = row0[j1c];
        b1[1] = row1[j1c];

        acc0 = __builtin_amdgcn_wmma_f32_16x16x4_f32(false, a, false, b0,
                                                     (short)0, acc0, false, false);
        acc1 = __builtin_amdgcn_wmma_f32_16x16x4_f32(false, a, false, b1,
                                                     (short)0, acc1, false, false);
    }

    // Row 0 of D lives in lanes 0..15, VGPR 0
    if (hi == 0) {
        if (j0 < N_MEM) sims[j0] = expf(acc0[0]);
        if (j1 < N_MEM) sims[j1] = expf(acc1[0]);
    }
}

// ---------------------------------------------------------------------------
// Kernel 3: p_partial[c] += sum_{j in chunk} sims[j] * V[j][c]
// 500 blocks x 400-row chunks; thread t (t<250) owns classes 4t..4t+3 and
// streams V rows with coalesced float4 loads (1000 floats = full row/block).
// ---------------------------------------------------------------------------
#define NB3 500
#define JCHUNK 400

__global__ __launch_bounds__(256) void pmem_kernel(const float* __restrict__ V,
                                                   const float* __restrict__ sims,
                                                   float* __restrict__ p_partial) {
    const int t = threadIdx.x;
    if (t >= 250) return;
    const int c = t * 4;
    const int jstart = blockIdx.x * JCHUNK;

    float ax = 0.0f, ay = 0.0f, az = 0.0f, aw = 0.0f;
    for (int j = jstart; j < jstart + JCHUNK; ++j) {
        const float s = sims[j];
        float4 vv = *(const float4*)(V + (size_t)j * N_CLASSES + c);
        ax = fmaf(s, vv.x, ax);
        ay = fmaf(s, vv.y, ay);
        az = fmaf(s, vv.z, az);
        aw = fmaf(s, vv.w, aw);
    }
    unsafeAtomicAdd(&p_partial[c + 0], ax);
    unsafeAtomicAdd(&p_partial[c + 1], ay);
    unsafeAtomicAdd(&p_partial[c + 2], az);
    unsafeAtomicAdd(&p_partial[c + 3], aw);
}

// ---------------------------------------------------------------------------
// Kernel 4: out = log(p)
// ---------------------------------------------------------------------------
__global__ __launch_bounds__(256) void log_kernel(const float* __restrict__ p_partial,
                                                  float* __restrict__ out) {
    const int c = blockIdx.x * 256 + threadIdx.x;
    if (c < N_CLASSES) out[c] = logf(p_partial[c]);
}

// ---------------------------------------------------------------------------
// launch
// ---------------------------------------------------------------------------
extern "C" void kernel_launch(void* const* d_in, const int* in_sizes, int n_in,
                              void* d_out, int out_size, void* d_ws, size_t ws_size,
                              hipStream_t stream) {
    (void)in_sizes; (void)n_in; (void)out_size; (void)ws_size;

    const float* x  = (const float*)d_in[0];   // [1, 2048]
    const float* K  = (const float*)d_in[1];   // [2048, 200000]
    const float* V  = (const float*)d_in[2];   // [200000, 1000]
    float*       out = (float*)d_out;          // [1, 1000]

    float* w         = (float*)d_ws;
    float* vn        = w;                      // 2048 floats
    float* p_partial = w + D_FEAT;             // 1024 floats (zeroed in prep)
    float* sims      = w + D_FEAT + 1024;      // 200000 floats

    prep_kernel<<<1, 256, 0, stream>>>(x, vn, p_partial);

    const int nblk2 = (N_MEM + 255) / 256;     // 256 columns per block
    sims_wmma_kernel<<<nblk2, 256, 0, stream>>>(K, vn, sims);

    pmem_kernel<<<NB3, 256, 0, stream>>>(V, sims, p_partial);

    log_kernel<<<(N_CLASSES + 255) / 256, 256, 0, stream>>>(p_partial, out);
}